// CenterAlignedTripletLoss_66992899883678
// MI455X (gfx1250) — compile-verified
//
#include <hip/hip_runtime.h>
#include <hip/hip_bf16.h>
#include <math.h>

typedef __attribute__((ext_vector_type(2))) float v2f;
typedef __attribute__((ext_vector_type(4))) float v4f;
typedef __attribute__((ext_vector_type(8))) float v8f;

#define M_TOT   1536   // rows of feats
#define N_CTR   96     // number of centers / anchors
#define KPG     16     // samples per center
#define FDIM    8192   // feature dim
#define NSTRIPE 8
#define DLOC    1024
#define MARGIN_F 0.3f
#define KSPLIT  8
#define KCHUNK  (FDIM / KSPLIT)   // 1024
#define GSIZE   (N_CTR * M_TOT)   // 147456

// ---------------- Kernel 1: centers = mean over 16 rows ----------------
__global__ void k_centers(const float* __restrict__ feats, float* __restrict__ centers) {
    int idx = blockIdx.x * blockDim.x + threadIdx.x;   // 0 .. 96*8192-1
    int n = idx >> 13;                                  // / 8192
    int d = idx & (FDIM - 1);
    const float* p = feats + (long)n * KPG * FDIM + d;
    float s = 0.f;
#pragma unroll
    for (int k = 0; k < KPG; ++k) s += p[(long)k * FDIM];
    centers[idx] = s * (1.0f / (float)KPG);
}

// ---------------- Kernel 2: row squared norms ----------------
__global__ void k_rownorm(const float* __restrict__ x, float* __restrict__ out, int cols) {
    __shared__ float sm[256];
    int row = blockIdx.x;
    const float* p = x + (long)row * cols;
    float s = 0.f;
    for (int d = threadIdx.x; d < cols; d += blockDim.x) { float v = p[d]; s += v * v; }
    sm[threadIdx.x] = s;
    __syncthreads();
    for (int off = 128; off > 0; off >>= 1) {
        if (threadIdx.x < off) sm[threadIdx.x] += sm[threadIdx.x + off];
        __syncthreads();
    }
    if (threadIdx.x == 0) out[row] = sm[0];
}

// ---------------- Kernel 3: K-split WMMA fp32 partial GEMM (centers @ feats^T) ----------
// 4608 waves: wave = (ks, mt, nt). Each computes a 16x16 partial tile over 1024 K.
// K-slot remap: lane-half h loads float4 at k+4h; 4 WMMAs consume .xy/.zw of two
// such vectors, covering 16 distinct k per step with B128 loads only.
__global__ void k_gemm_part(const float* __restrict__ centers,
                            const float* __restrict__ feats,
                            float* __restrict__ gpart) {
    int w    = (blockIdx.x * blockDim.x + threadIdx.x) >> 5;  // 0..4607
    int lane = threadIdx.x & 31;
    int ks = w & (KSPLIT - 1);
    int t  = w >> 3;                    // 0..575
    int mt = t % (N_CTR / 16);          // 0..5
    int nt = t / (N_CTR / 16);          // 0..95

    int rl = lane & 15;                 // A row / B col within tile
    int h  = lane >> 4;                 // lane half
    int k0 = ks * KCHUNK + 4 * h;

    const float* pa = centers + (long)(mt * 16 + rl) * FDIM + k0;
    const float* pb = feats   + (long)(nt * 16 + rl) * FDIM + k0;

    v8f acc = {};
#pragma unroll 2
    for (int k = 0; k < KCHUNK; k += 16) {
        v4f a0 = *(const v4f*)(pa + k);
        v4f a1 = *(const v4f*)(pa + k + 8);
        v4f b0 = *(const v4f*)(pb + k);
        v4f b1 = *(const v4f*)(pb + k + 8);
        acc = __builtin_amdgcn_wmma_f32_16x16x4_f32(false, __builtin_shufflevector(a0, a0, 0, 1),
                                                    false, __builtin_shufflevector(b0, b0, 0, 1),
                                                    (short)0, acc, false, false);
        acc = __builtin_amdgcn_wmma_f32_16x16x4_f32(false, __builtin_shufflevector(a0, a0, 2, 3),
                                                    false, __builtin_shufflevector(b0, b0, 2, 3),
                                                    (short)0, acc, false, false);
        acc = __builtin_amdgcn_wmma_f32_16x16x4_f32(false, __builtin_shufflevector(a1, a1, 0, 1),
                                                    false, __builtin_shufflevector(b1, b1, 0, 1),
                                                    (short)0, acc, false, false);
        acc = __builtin_amdgcn_wmma_f32_16x16x4_f32(false, __builtin_shufflevector(a1, a1, 2, 3),
                                                    false, __builtin_shufflevector(b1, b1, 2, 3),
                                                    (short)0, acc, false, false);
    }

    // C/D layout: VGPR r -> row (r + 8*h), col = lane&15
    int m0 = mt * 16 + (h << 3);
    int n  = nt * 16 + rl;
    float* out = gpart + (long)ks * GSIZE;
#pragma unroll
    for (int r = 0; r < 8; ++r)
        out[(long)(m0 + r) * M_TOT + n] = acc[r];
}

// ---------------- Kernel 3b: fold K-split partials + distance ----------------
__global__ void k_reduce_dist(const float* __restrict__ gpart,
                              const float* __restrict__ cn, const float* __restrict__ fn,
                              float* __restrict__ dist) {
    int idx = blockIdx.x * blockDim.x + threadIdx.x;   // 0..147455
    int m = idx / M_TOT;
    int n = idx - m * M_TOT;
    float s = 0.f;
#pragma unroll
    for (int ks = 0; ks < KSPLIT; ++ks) s += gpart[(long)ks * GSIZE + idx];
    float d2 = cn[m] + fn[n] - 2.0f * s;
    dist[idx] = sqrtf(fmaxf(d2, 1e-12f));
}

// ---------------- Kernel 4: masked argmax (pos) / argmin (neg) per anchor ----------------
__global__ void k_select(const float* __restrict__ dist, const int* __restrict__ labels,
                         int* __restrict__ pind, int* __restrict__ nind) {
    __shared__ float spv[256]; __shared__ int spi[256];
    __shared__ float snv[256]; __shared__ int sni[256];
    int i  = blockIdx.x;
    int al = labels[i * KPG];
    float pv = -INFINITY; int pi = 0x7fffffff;
    float nv =  INFINITY; int ni = 0x7fffffff;
    const float* drow = dist + (long)i * M_TOT;
    for (int j = threadIdx.x; j < M_TOT; j += blockDim.x) {
        float v = drow[j];
        if (labels[j] == al) {
            if (v > pv || (v == pv && j < pi)) { pv = v; pi = j; }
        } else {
            if (v < nv || (v == nv && j < ni)) { nv = v; ni = j; }
        }
    }
    spv[threadIdx.x] = pv; spi[threadIdx.x] = pi;
    snv[threadIdx.x] = nv; sni[threadIdx.x] = ni;
    __syncthreads();
    for (int off = 128; off > 0; off >>= 1) {
        if (threadIdx.x < off) {
            int o = threadIdx.x + off;
            if (spv[o] > spv[threadIdx.x] || (spv[o] == spv[threadIdx.x] && spi[o] < spi[threadIdx.x])) {
                spv[threadIdx.x] = spv[o]; spi[threadIdx.x] = spi[o];
            }
            if (snv[o] < snv[threadIdx.x] || (snv[o] == snv[threadIdx.x] && sni[o] < sni[threadIdx.x])) {
                snv[threadIdx.x] = snv[o]; sni[threadIdx.x] = sni[o];
            }
        }
        __syncthreads();
    }
    if (threadIdx.x == 0) { pind[i] = spi[0]; nind[i] = sni[0]; }
}

// ---------------- Kernel 5: local stripe distance + 8x8 shortest-path DP ----------------
// blockIdx.x = 2*anchor + which (0 = positive, 1 = negative)
__global__ void k_localdist(const float* __restrict__ centers,
                            const float* __restrict__ lfeat,
                            const int* __restrict__ pind, const int* __restrict__ nind,
                            float* __restrict__ ap_out, float* __restrict__ an_out) {
    __shared__ float ys[NSTRIPE * DLOC];   // transposed gathered local features (32 KB)
    __shared__ float dmat[64];
    int i     = blockIdx.x >> 1;
    int which = blockIdx.x & 1;
    int ind   = which ? nind[i] : pind[i];

    // y[s][d] = local_features[ind][d][s]  (lfeat flat: m*8192 + d*8 + s)
    for (int t = threadIdx.x; t < NSTRIPE * DLOC; t += blockDim.x) {
        int d = t >> 3, s = t & 7;
        ys[s * DLOC + d] = lfeat[(long)ind * FDIM + t];
    }
    __syncthreads();

    if (threadIdx.x < 64) {
        int a = threadIdx.x >> 3, b = threadIdx.x & 7;
        const float* xa = centers + (long)i * FDIM + a * DLOC;  // inputs1[i][a][:]
        const float* yb = ys + b * DLOC;
        float xn = 0.f, yn = 0.f, xy = 0.f;
        for (int d = 0; d < DLOC; ++d) {
            float xv = xa[d], yv = yb[d];
            xn += xv * xv; yn += yv * yv; xy += xv * yv;
        }
        float d2 = xn + yn - 2.0f * xy;
        float dd = sqrtf(fmaxf(d2, 1e-12f));
        float e  = expf(dd);
        dmat[threadIdx.x] = (e - 1.0f) / (e + 1.0f);
    }
    __syncthreads();

    if (threadIdx.x == 0) {
        float dp[NSTRIPE][NSTRIPE];
        for (int a = 0; a < NSTRIPE; ++a)
            for (int b = 0; b < NSTRIPE; ++b) {
                float v = dmat[a * 8 + b];
                if (a == 0 && b == 0)      dp[a][b] = v;
                else if (a == 0)           dp[a][b] = dp[a][b - 1] + v;
                else if (b == 0)           dp[a][b] = dp[a - 1][b] + v;
                else                       dp[a][b] = fminf(dp[a - 1][b], dp[a][b - 1]) + v;
            }
        float* out = which ? an_out : ap_out;
        out[i] = dp[NSTRIPE - 1][NSTRIPE - 1];
    }
}

// ---------------- Kernel 6: final loss ----------------
__global__ void k_loss(const float* __restrict__ ap, const float* __restrict__ an,
                       float* __restrict__ out) {
    __shared__ float sm[128];
    float s = 0.f;
    for (int i = threadIdx.x; i < N_CTR; i += blockDim.x)
        s += fmaxf(ap[i] - an[i] + MARGIN_F, 0.0f);
    sm[threadIdx.x] = s;
    __syncthreads();
    for (int off = 64; off > 0; off >>= 1) {
        if (threadIdx.x < off) sm[threadIdx.x] += sm[threadIdx.x + off];
        __syncthreads();
    }
    if (threadIdx.x == 0) out[0] = sm[0] / (float)N_CTR;
}

extern "C" void kernel_launch(void* const* d_in, const int* in_sizes, int n_in,
                              void* d_out, int out_size, void* d_ws, size_t ws_size,
                              hipStream_t stream) {
    const float* feats  = (const float*)d_in[0];   // (1536, 8192) f32
    const int*   labels = (const int*)d_in[1];     // (1536,) i32
    const float* lfeat  = (const float*)d_in[2];   // (1536, 1024, 8) f32
    float* out = (float*)d_out;

    // Workspace layout (floats)
    float* W       = (float*)d_ws;
    float* centers = W;                       // 96*8192          = 786432
    float* gpart   = W + 786432;              // 8*96*1536        = 1179648
    float* dist    = W + 1966080;             // 96*1536          = 147456
    float* cn      = W + 2113536;             // 96
    float* fn      = W + 2113632;             // 1536
    float* apv     = W + 2115168;             // 96
    float* anv     = W + 2115264;             // 96
    int*   pind    = (int*)(W + 2115360);     // 96
    int*   nind    = pind + 96;               // 96   (total ~8.5 MB)

    k_centers<<<(N_CTR * FDIM) / 256, 256, 0, stream>>>(feats, centers);
    k_rownorm<<<N_CTR, 256, 0, stream>>>(centers, cn, FDIM);
    k_rownorm<<<M_TOT, 256, 0, stream>>>(feats, fn, FDIM);
    k_gemm_part<<<(KSPLIT * 576 * 32) / 256, 256, 0, stream>>>(centers, feats, gpart);
    k_reduce_dist<<<GSIZE / 256, 256, 0, stream>>>(gpart, cn, fn, dist);
    k_select<<<N_CTR, 256, 0, stream>>>(dist, labels, pind, nind);
    k_localdist<<<2 * N_CTR, 256, 0, stream>>>(centers, lfeat, pind, nind, apv, anv);
    k_loss<<<1, 128, 0, stream>>>(apv, anv, out);
}